// GATModelEncoderStatic_46591805227654
// MI455X (gfx1250) — compile-verified
//
#include <hip/hip_runtime.h>
#include <math.h>

// Problem constants (match reference)
#define N_NODES   50000
#define N_EDGES   800000
#define ETOT      (N_EDGES + N_NODES)   // with self-loops
#define IN_CH     128
#define HID       64
#define HEADS     4
#define OUT_CH    64
#define HC1       (HEADS * HID)         // 256
#define NEG_SLOPE 0.2f

typedef float v2f __attribute__((ext_vector_type(2)));
typedef float v8f __attribute__((ext_vector_type(8)));

// ---------------------------------------------------------------------------
// Fill helpers (graph-capture safe; no hipMemset)
// ---------------------------------------------------------------------------
__global__ void fill_bcast_kernel(float* __restrict__ dst, const float* __restrict__ src,
                                  int len, int period) {
  int i = blockIdx.x * blockDim.x + threadIdx.x;
  if (i < len) dst[i] = src[i % period];
}

__global__ void fill_val_kernel(float* __restrict__ dst, float v, int len) {
  int i = blockIdx.x * blockDim.x + threadIdx.x;
  if (i < len) dst[i] = v;
}

// ---------------------------------------------------------------------------
// f32 WMMA GEMM: C[M,Nc] = A[M,K] * B[K,Nc], row-major, one wave per 16x16 tile.
// V_WMMA_F32_16X16X4_F32: A is 16x4, B is 4x16, C/D 16x16 f32 (8 VGPRs).
// A layout  (32-bit, 16x4): lane L -> row M = L&15; a[v] = A[M][k0 + 2*(L>>4) + v]
// B layout  (32-bit, 4x16): lane L -> col N = L&15; b[v] = B[k0 + 2*(L>>4) + v][N]
// C/D layout: lane L, vgpr v -> M = v + 8*(L>>4), N = L&15
// M, Nc multiples of 16; K multiple of 4 (true for all shapes here).
// ---------------------------------------------------------------------------
__global__ __launch_bounds__(32) void gemm_wmma_f32(const float* __restrict__ A,
                                                    const float* __restrict__ B,
                                                    float* __restrict__ C,
                                                    int M, int K, int Nc) {
  const int lane = threadIdx.x & 31;
  const int grp  = lane >> 4;     // 0 or 1 (K half for A/B, M half for C/D)
  const int mn   = lane & 15;     // row of A / col of B / col of C
  const int tm   = blockIdx.y * 16;
  const int tn   = blockIdx.x * 16;

  const float* __restrict__ arow = A + (size_t)(tm + mn) * K;
  const float* __restrict__ bcol = B + (tn + mn);

  v8f acc = {0.f, 0.f, 0.f, 0.f, 0.f, 0.f, 0.f, 0.f};
  for (int k0 = 0; k0 < K; k0 += 4) {
    const int ka = k0 + 2 * grp;
    v2f a, b;
    a.x = arow[ka];
    a.y = arow[ka + 1];
    b.x = bcol[(size_t)ka * Nc];
    b.y = bcol[(size_t)(ka + 1) * Nc];
    acc = __builtin_amdgcn_wmma_f32_16x16x4_f32(false, a, false, b,
                                                (short)0, acc, false, false);
  }

  float* __restrict__ crow = C + (size_t)(tm + 8 * grp) * Nc + tn + mn;
#pragma unroll
  for (int v = 0; v < 8; ++v) crow[(size_t)v * Nc] = acc[v];
}

// ---------------------------------------------------------------------------
// Per-node attention scores: alpha_s/d[n,h] = dot(h[n, h*C : h*C+C], a_src/dst[h])
// ---------------------------------------------------------------------------
__global__ void scores_kernel(const float* __restrict__ h,
                              const float* __restrict__ a_src,
                              const float* __restrict__ a_dst,
                              float* __restrict__ alpha_s,
                              float* __restrict__ alpha_d,
                              int n_nodes, int heads, int ch) {
  int idx = blockIdx.x * blockDim.x + threadIdx.x;
  if (idx >= n_nodes * heads) return;
  int n = idx / heads, hd = idx % heads;
  const float* hp = h + (size_t)n * heads * ch + (size_t)hd * ch;
  const float* as = a_src + hd * ch;
  const float* ad = a_dst + hd * ch;
  float ss = 0.f, sd = 0.f;
  for (int c = 0; c < ch; ++c) {
    float v = hp[c];
    ss += v * as[c];
    sd += v * ad[c];
  }
  alpha_s[idx] = ss;
  alpha_d[idx] = sd;
}

// Sign-split float atomic max (works for mixed-sign with -inf init).
__device__ inline void atomicMaxF(float* addr, float val) {
  if (val >= 0.f)
    atomicMax((int*)addr, __float_as_int(val));
  else
    atomicMin((unsigned int*)addr, __float_as_uint(val));
}

__device__ inline void edge_endpoints(const int* __restrict__ ei, int e,
                                      int n_edges, int& s, int& d) {
  if (e < n_edges) { s = ei[e]; d = ei[n_edges + e]; }
  else             { s = d = e - n_edges; }           // self-loop
}

// Pass 1: leaky-relu(alpha_s[src]+alpha_d[dst]) -> w buffer; atomic max into m[dst].
__global__ void edge_alpha_max_kernel(const int* __restrict__ ei,
                                      const float* __restrict__ alpha_s,
                                      const float* __restrict__ alpha_d,
                                      float* __restrict__ m,
                                      float* __restrict__ w,
                                      int n_edges, int n_nodes, int heads) {
  int e = blockIdx.x * blockDim.x + threadIdx.x;
  if (e >= n_edges + n_nodes) return;
  int s, d;
  edge_endpoints(ei, e, n_edges, s, d);
  for (int hd = 0; hd < heads; ++hd) {
    float a = alpha_s[s * heads + hd] + alpha_d[d * heads + hd];
    a = (a > 0.f) ? a : NEG_SLOPE * a;
    w[(size_t)e * heads + hd] = a;
    atomicMaxF(&m[d * heads + hd], a);
  }
}

// Pass 2: denom[dst] += exp(alpha - m[dst])
__global__ void edge_expsum_kernel(const int* __restrict__ ei,
                                   const float* __restrict__ w,
                                   const float* __restrict__ m,
                                   float* __restrict__ denom,
                                   int n_edges, int n_nodes, int heads) {
  int e = blockIdx.x * blockDim.x + threadIdx.x;
  if (e >= n_edges + n_nodes) return;
  int s, d;
  edge_endpoints(ei, e, n_edges, s, d);
  for (int hd = 0; hd < heads; ++hd) {
    float a  = w[(size_t)e * heads + hd];
    float ev = __expf(a - m[d * heads + hd]) ;
    ev = expf(a - m[d * heads + hd]);   // precise exp (overwrite fast approx)
    atomicAdd(&denom[d * heads + hd], ev);
  }
}

// Pass 3: w[e,h] = exp(alpha - m[dst]) / (denom[dst] + 1e-16)
__global__ void edge_weight_kernel(const int* __restrict__ ei,
                                   float* __restrict__ w,
                                   const float* __restrict__ m,
                                   const float* __restrict__ denom,
                                   int n_edges, int n_nodes, int heads) {
  int e = blockIdx.x * blockDim.x + threadIdx.x;
  if (e >= n_edges + n_nodes) return;
  int s, d;
  edge_endpoints(ei, e, n_edges, s, d);
  for (int hd = 0; hd < heads; ++hd) {
    float a = w[(size_t)e * heads + hd];
    float ev = expf(a - m[d * heads + hd]);
    w[(size_t)e * heads + hd] = ev / (denom[d * heads + hd] + 1e-16f);
  }
}

// Pass 4: out[dst, :] += w[e, head(c)] * h[src, :]   (one wave per edge)
__global__ __launch_bounds__(256) void aggregate_kernel(const int* __restrict__ ei,
                                                        const float* __restrict__ h,
                                                        const float* __restrict__ w,
                                                        float* __restrict__ out,
                                                        int n_edges, int n_nodes,
                                                        int heads, int ch) {
  const int lane = threadIdx.x & 31;
  const long wid = (long)blockIdx.x * (blockDim.x >> 5) + (threadIdx.x >> 5);
  if (wid >= (long)(n_edges + n_nodes)) return;
  int e = (int)wid;
  int s, d;
  edge_endpoints(ei, e, n_edges, s, d);
  const int hc = heads * ch;
  const float* __restrict__ hs = h + (size_t)s * hc;
  float* __restrict__ od       = out + (size_t)d * hc;
  const float* __restrict__ we = w + (size_t)e * heads;
  for (int c = lane; c < hc; c += 32) {
    atomicAdd(&od[c], we[c / ch] * hs[c]);   // coalesced gather + L2 atomic scatter
  }
}

__global__ void elu_inplace_kernel(float* __restrict__ x, int len) {
  int i = blockIdx.x * blockDim.x + threadIdx.x;
  if (i < len) {
    float v = x[i];
    x[i] = (v > 0.f) ? v : expm1f(v);
  }
}

// ---------------------------------------------------------------------------
// Launcher
// ---------------------------------------------------------------------------
extern "C" void kernel_launch(void* const* d_in, const int* in_sizes, int n_in,
                              void* d_out, int out_size, void* d_ws, size_t ws_size,
                              hipStream_t stream) {
  const float* x   = (const float*)d_in[0];
  const int*   ei  = (const int*)  d_in[1];   // [2, E]
  const float* W1  = (const float*)d_in[2];   // [128, 256]
  const float* as1 = (const float*)d_in[3];   // [4, 64]
  const float* ad1 = (const float*)d_in[4];
  const float* b1  = (const float*)d_in[5];   // [256]
  const float* W2  = (const float*)d_in[6];   // [256, 64]
  const float* as2 = (const float*)d_in[7];   // [1, 64]
  const float* ad2 = (const float*)d_in[8];
  const float* b2  = (const float*)d_in[9];   // [64]
  float* out = (float*)d_out;                 // [N, 64]

  // Workspace layout (floats)
  float* p       = (float*)d_ws;
  float* h1      = p; p += (size_t)N_NODES * HC1;    // layer1 features
  float* agg1    = p; p += (size_t)N_NODES * HC1;    // layer1 aggregation / layer2 input
  float* h2      = p; p += (size_t)N_NODES * OUT_CH; // layer2 features
  float* alpha_s = p; p += (size_t)N_NODES * HEADS;
  float* alpha_d = p; p += (size_t)N_NODES * HEADS;
  float* mbuf    = p; p += (size_t)N_NODES * HEADS;
  float* denom   = p; p += (size_t)N_NODES * HEADS;
  float* wbuf    = p; p += (size_t)ETOT    * HEADS;

  const int TB = 256;
  const int edge_blocks = (ETOT + TB - 1) / TB;
  const int wave_blocks = (ETOT + 7) / 8;            // 8 waves per 256-thread block

  // ---------------- Layer 1 (HEADS=4, HID=64) ----------------
  gemm_wmma_f32<<<dim3(HC1 / 16, N_NODES / 16), 32, 0, stream>>>(x, W1, h1,
                                                                 N_NODES, IN_CH, HC1);
  {
    int nh = N_NODES * HEADS;
    scores_kernel<<<(nh + TB - 1) / TB, TB, 0, stream>>>(h1, as1, ad1,
                                                         alpha_s, alpha_d,
                                                         N_NODES, HEADS, HID);
    fill_val_kernel<<<(nh + TB - 1) / TB, TB, 0, stream>>>(mbuf, -INFINITY, nh);
    fill_val_kernel<<<(nh + TB - 1) / TB, TB, 0, stream>>>(denom, 0.f, nh);
    int len = N_NODES * HC1;
    fill_bcast_kernel<<<(len + TB - 1) / TB, TB, 0, stream>>>(agg1, b1, len, HC1);

    edge_alpha_max_kernel<<<edge_blocks, TB, 0, stream>>>(ei, alpha_s, alpha_d,
                                                          mbuf, wbuf,
                                                          N_EDGES, N_NODES, HEADS);
    edge_expsum_kernel<<<edge_blocks, TB, 0, stream>>>(ei, wbuf, mbuf, denom,
                                                       N_EDGES, N_NODES, HEADS);
    edge_weight_kernel<<<edge_blocks, TB, 0, stream>>>(ei, wbuf, mbuf, denom,
                                                       N_EDGES, N_NODES, HEADS);
    aggregate_kernel<<<wave_blocks, TB, 0, stream>>>(ei, h1, wbuf, agg1,
                                                     N_EDGES, N_NODES, HEADS, HID);
    elu_inplace_kernel<<<(len + TB - 1) / TB, TB, 0, stream>>>(agg1, len);
  }

  // ---------------- Layer 2 (heads=1, OUT=64) ----------------
  gemm_wmma_f32<<<dim3(OUT_CH / 16, N_NODES / 16), 32, 0, stream>>>(agg1, W2, h2,
                                                                    N_NODES, HC1, OUT_CH);
  {
    int nh = N_NODES;  // heads = 1
    scores_kernel<<<(nh + TB - 1) / TB, TB, 0, stream>>>(h2, as2, ad2,
                                                         alpha_s, alpha_d,
                                                         N_NODES, 1, OUT_CH);
    fill_val_kernel<<<(nh + TB - 1) / TB, TB, 0, stream>>>(mbuf, -INFINITY, nh);
    fill_val_kernel<<<(nh + TB - 1) / TB, TB, 0, stream>>>(denom, 0.f, nh);
    int len = N_NODES * OUT_CH;
    fill_bcast_kernel<<<(len + TB - 1) / TB, TB, 0, stream>>>(out, b2, len, OUT_CH);

    edge_alpha_max_kernel<<<edge_blocks, TB, 0, stream>>>(ei, alpha_s, alpha_d,
                                                          mbuf, wbuf,
                                                          N_EDGES, N_NODES, 1);
    edge_expsum_kernel<<<edge_blocks, TB, 0, stream>>>(ei, wbuf, mbuf, denom,
                                                       N_EDGES, N_NODES, 1);
    edge_weight_kernel<<<edge_blocks, TB, 0, stream>>>(ei, wbuf, mbuf, denom,
                                                       N_EDGES, N_NODES, 1);
    aggregate_kernel<<<wave_blocks, TB, 0, stream>>>(ei, h2, wbuf, out,
                                                     N_EDGES, N_NODES, 1, OUT_CH);
  }
}